// SSD300_ext_heads_7524782702807
// MI455X (gfx1250) — compile-verified
//
#include <hip/hip_runtime.h>

// ---------------------------------------------------------------------------
// SSD300 extra heads on MI455X (gfx1250), fp32 via V_WMMA_F32_16X16X4_F32.
// Implicit GEMM: 3x3 SAME conv = sum of 9 shifted 1x1 GEMMs.
//   M = Cout, N = B*H*W (B=16), K = 256*9.
// Workgroup = 256 threads (8 wave32). Tile: M=128 x N=64.
// Wave w: co rows [w*16, w*16+16), 4 WMMA accum tiles covering N=64.
// LDS: 64x32 position-major im2col slice + 128x32 weight slice, stride 34
// (conflict-free staging stores and v2f fragment reads; even stride keeps
// 8-byte alignment for the 2-dword A/B fragments).
// Round 2: all division/address math hoisted out of the tap/chunk loops —
// each thread stages one fixed spatial position (low 6 bits of idx are
// invariant across the staging sub-iterations), so (b,h,w), validity and
// base addresses are computed once and advanced affinely.
// ---------------------------------------------------------------------------

typedef float v2f __attribute__((ext_vector_type(2)));
typedef float v8f __attribute__((ext_vector_type(8)));

#define CIN   256
#define TB    16
#define BSTR  34
#define WSTR  34
#define TOTAL_ANCH 11640

__global__ __launch_bounds__(256)
void conv3x3_mid_wmma(const float* __restrict__ X,
                      const float* __restrict__ Wt,
                      const float* __restrict__ Bias,
                      float* __restrict__ Y,
                      int H, int W)
{
    __shared__ __align__(16) float sB[64 * BSTR];
    __shared__ __align__(16) float sW[128 * WSTR];

    const int HW   = H * W;
    const int N    = TB * HW;
    const int tid  = threadIdx.x;
    const int lane = tid & 31;
    const int wave = tid >> 5;
    const int m    = lane & 15;          // row/col within 16
    const int ksel = (lane >> 4) << 1;   // A/B K-pair: lanes 0-15 -> K{0,1}, 16-31 -> K{2,3}
    const int mo   = (lane >> 4) << 3;   // C/D: lanes 16-31 hold M+8
    const int n0   = blockIdx.x * 64;
    const int co_base = blockIdx.y * 128;
    const int co0  = co_base + wave * 16;

    // ---- hoisted staging geometry (one spatial position per thread) ----
    const int nl   = tid & 63;           // staged position, invariant across its
    const int cib  = tid >> 6;           // channel base 0..3, +4 per it
    const int p    = n0 + nl;
    const bool pval = p < N;
    const int pc   = pval ? p : 0;
    const int b_   = pc / HW;
    const int hw0  = pc - b_ * HW;
    const int h_   = hw0 / W;
    const int w_   = hw0 - h_ * W;
    const int gbase  = (b_ * CIN + cib) * HW + h_ * W + w_;  // + it*4*HW + ci0*HW + shift
    const int sbbase = nl * BSTR + cib;                       // + it*4

    const int wci  = tid & 31;           // weight ci, invariant across its
    const int wcol = tid >> 5;           // weight row base, +8 per it
    const int wbase  = ((co_base + wcol) * CIN + wci) * 9;    // + it*8*CIN*9 + ci0*9 + kpos
    const int swbase = wcol * WSTR + wci;                     // + it*8*WSTR

    v8f acc[4] = {};

    for (int kpos = 0; kpos < 9; ++kpos) {
        const int dy = kpos / 3 - 1;
        const int dx = kpos % 3 - 1;
        const int hh = h_ + dy, ww = w_ + dx;
        const bool ok = pval & ((unsigned)hh < (unsigned)H) & ((unsigned)ww < (unsigned)W);
        const int shift = dy * W + dx;

        for (int ci0 = 0; ci0 < CIN; ci0 += 32) {
            __syncthreads();
            // ---- stage 64x32 im2col slice: 1 select + 1 load + 1 ds_store per elem
            const float* Xp = X + gbase + ci0 * HW + shift;
            #pragma unroll
            for (int it = 0; it < 8; ++it) {
                float v = ok ? Xp[it * 4 * HW] : 0.0f;
                sB[sbbase + it * 4] = v;
            }
            // ---- stage 128x32 weight slice for this tap
            const float* Wp = Wt + wbase + ci0 * 9 + kpos;
            #pragma unroll
            for (int it = 0; it < 16; ++it)
                sW[swbase + it * 8 * WSTR] = Wp[it * 8 * CIN * 9];
            __syncthreads();

            const int arow = (wave * 16 + m) * WSTR + ksel;
            #pragma unroll
            for (int c4 = 0; c4 < 32; c4 += 4) {
                v2f a = *(const v2f*)&sW[arow + c4];
                #pragma unroll
                for (int t = 0; t < 4; ++t) {
                    v2f b = *(const v2f*)&sB[(t * 16 + m) * BSTR + c4 + ksel];
                    acc[t] = __builtin_amdgcn_wmma_f32_16x16x4_f32(
                        false, a, false, b, (short)0, acc[t], false, false);
                }
            }
        }
    }

    // ---- epilogue: bias + ReLU, NCHW store
    #pragma unroll
    for (int t = 0; t < 4; ++t) {
        int pp = n0 + t * 16 + m;
        if (pp >= N) continue;
        int bb = pp / HW;
        int hw = pp - bb * HW;
        #pragma unroll
        for (int r = 0; r < 8; ++r) {
            int co = co0 + mo + r;
            float v = acc[t][r] + Bias[co];
            Y[((long)bb * CIN + co) * HW + hw] = fmaxf(v, 0.0f);
        }
    }
}

// Final layer: Cout = 24 (reg) or 54 (cls). Scatters straight into the
// reshape(B,nq,-1)+concat output: channel c -> q=c/6, anchor=c%6.
__global__ __launch_bounds__(256)
void conv3x3_final_wmma(const float* __restrict__ X,
                        const float* __restrict__ Wt,
                        const float* __restrict__ Bias,
                        float* __restrict__ Out,
                        int H, int W, int Cout, int fm_off, int nq)
{
    __shared__ __align__(16) float sB[64 * BSTR];
    __shared__ __align__(16) float sW[64 * WSTR];

    const int HW   = H * W;
    const int N    = TB * HW;
    const int tid  = threadIdx.x;
    const int lane = tid & 31;
    const int wave = tid >> 5;
    const int m    = lane & 15;
    const int ksel = (lane >> 4) << 1;
    const int mo   = (lane >> 4) << 3;
    const int n0   = blockIdx.x * 64;
    const int nActive = (Cout + 15) >> 4;   // 2 (reg) or 4 (cls) active waves

    const int nl   = tid & 63;
    const int cib  = tid >> 6;
    const int p    = n0 + nl;
    const bool pval = p < N;
    const int pc   = pval ? p : 0;
    const int b_   = pc / HW;
    const int hw0  = pc - b_ * HW;
    const int h_   = hw0 / W;
    const int w_   = hw0 - h_ * W;
    const int gbase  = (b_ * CIN + cib) * HW + h_ * W + w_;
    const int sbbase = nl * BSTR + cib;

    const int wci  = tid & 31;
    const int wcol = tid >> 5;               // 0..7, +8 per it -> rows 0..63
    const int wbase  = (wcol * CIN + wci) * 9;
    const int swbase = wcol * WSTR + wci;

    v8f acc[4] = {};

    for (int kpos = 0; kpos < 9; ++kpos) {
        const int dy = kpos / 3 - 1;
        const int dx = kpos % 3 - 1;
        const int hh = h_ + dy, ww = w_ + dx;
        const bool ok = pval & ((unsigned)hh < (unsigned)H) & ((unsigned)ww < (unsigned)W);
        const int shift = dy * W + dx;

        for (int ci0 = 0; ci0 < CIN; ci0 += 32) {
            __syncthreads();
            const float* Xp = X + gbase + ci0 * HW + shift;
            #pragma unroll
            for (int it = 0; it < 8; ++it) {
                float v = ok ? Xp[it * 4 * HW] : 0.0f;
                sB[sbbase + it * 4] = v;
            }
            const float* Wp = Wt + wbase + ci0 * 9 + kpos;
            #pragma unroll
            for (int it = 0; it < 8; ++it) {
                int col = wcol + it * 8;     // zero-pad rows beyond Cout
                float v = (col < Cout) ? Wp[it * 8 * CIN * 9] : 0.0f;
                sW[swbase + it * 8 * WSTR] = v;
            }
            __syncthreads();

            if (wave < nActive) {            // wave-uniform; EXEC stays all-ones
                const int arow = (wave * 16 + m) * WSTR + ksel;
                #pragma unroll
                for (int c4 = 0; c4 < 32; c4 += 4) {
                    v2f a = *(const v2f*)&sW[arow + c4];
                    #pragma unroll
                    for (int t = 0; t < 4; ++t) {
                        v2f b = *(const v2f*)&sB[(t * 16 + m) * BSTR + c4 + ksel];
                        acc[t] = __builtin_amdgcn_wmma_f32_16x16x4_f32(
                            false, a, false, b, (short)0, acc[t], false, false);
                    }
                }
            }
        }
    }

    if (wave < nActive) {
        #pragma unroll
        for (int t = 0; t < 4; ++t) {
            int pp = n0 + t * 16 + m;
            if (pp >= N) continue;
            int bb = pp / HW;
            int hw = pp - bb * HW;
            #pragma unroll
            for (int r = 0; r < 8; ++r) {
                int c = wave * 16 + mo + r;
                if (c < Cout) {
                    int q = c / 6;
                    int a = c - q * 6;
                    Out[((long)bb * nq + q) * TOTAL_ANCH + fm_off + a * HW + hw] =
                        acc[t][r] + Bias[c];
                }
            }
        }
    }
}

extern "C" void kernel_launch(void* const* d_in, const int* in_sizes, int n_in,
                              void* d_out, int out_size, void* d_ws, size_t ws_size,
                              hipStream_t stream)
{
    (void)in_sizes; (void)n_in; (void)out_size; (void)ws_size;

    const float* feats[6];
    for (int i = 0; i < 6; ++i) feats[i] = (const float*)d_in[i];
    // setup_inputs dict order: rw0,rb0,rw1,rb1,rw2,rb2,rw3,rb3,rw4,rb4, then cls.
    const float* rw[5] = {(const float*)d_in[6],  (const float*)d_in[8],
                          (const float*)d_in[10], (const float*)d_in[12],
                          (const float*)d_in[14]};
    const float* rb[5] = {(const float*)d_in[7],  (const float*)d_in[9],
                          (const float*)d_in[11], (const float*)d_in[13],
                          (const float*)d_in[15]};
    const float* cw[5] = {(const float*)d_in[16], (const float*)d_in[18],
                          (const float*)d_in[20], (const float*)d_in[22],
                          (const float*)d_in[24]};
    const float* cb[5] = {(const float*)d_in[17], (const float*)d_in[19],
                          (const float*)d_in[21], (const float*)d_in[23],
                          (const float*)d_in[25]};

    static const int  FH[6]      = {38, 19, 10, 5, 3, 1};
    static const long bufOff[6]  = {0, 5914624, 7393280, 7802880, 7905280, 7942144};
    static const int  anchOff[6] = {0, 8664, 10830, 11430, 11580, 11634};

    float* bufA = (float*)d_ws;                  // 31.8 MB ping
    float* bufB = bufA + 7946240;                // 31.8 MB pong
    float* outReg = (float*)d_out;                               // [16,4,11640]
    float* outCls = outReg + (long)TB * 4 * TOTAL_ANCH;          // [16,9,11640]

    for (int head = 0; head < 2; ++head) {
        const float* const* ww = head ? cw : rw;
        const float* const* bb = head ? cb : rb;
        float* outp = head ? outCls : outReg;
        const int Cout = head ? 54 : 24;
        const int nq   = head ? 9 : 4;
        for (int fm = 0; fm < 6; ++fm) {
            const int H = FH[fm];
            const int N = TB * H * H;
            dim3 block(256);
            dim3 grid((N + 63) / 64, 2);     // Cout=256 -> 2 M-tiles of 128
            float* pa = bufA + bufOff[fm];
            float* pb = bufB + bufOff[fm];
            conv3x3_mid_wmma<<<grid, block, 0, stream>>>(feats[fm], ww[0], bb[0], pa, H, H);
            conv3x3_mid_wmma<<<grid, block, 0, stream>>>(pa,        ww[1], bb[1], pb, H, H);
            conv3x3_mid_wmma<<<grid, block, 0, stream>>>(pb,        ww[2], bb[2], pa, H, H);
            conv3x3_mid_wmma<<<grid, block, 0, stream>>>(pa,        ww[3], bb[3], pb, H, H);
            dim3 gridf((N + 63) / 64, 1);
            conv3x3_final_wmma<<<gridf, block, 0, stream>>>(pb, ww[4], bb[4], outp,
                                                            H, H, Cout, anchOff[fm], nq);
        }
    }
}